// EfficientEmbeddingLayer_37864431681724
// MI455X (gfx1250) — compile-verified
//
#include <hip/hip_runtime.h>

typedef __attribute__((ext_vector_type(2))) float v2f;
typedef __attribute__((ext_vector_type(8))) float v8f;

#define NUM_TOKENS 16384   // BATCH * SEQ = 8 * 2048
#define RANK       256
#define EMB_DIM    1024
#define M_BLK      64      // tokens per block (4 x 16-row WMMA subtiles)
#define N_BLK      256     // output columns per block (4 waves x 64)
#define LDS_STRIDE 260     // RANK + 4 floats pad -> conflict-free ds_load_b64

__global__ __launch_bounds__(256, 1)
void factored_embed_wmma(const int* __restrict__ x,
                         const float* __restrict__ A,
                         const float* __restrict__ B,
                         float* __restrict__ out)
{
    // Staged gathered A rows: 64 rows x 260 floats = 66,560 B (of 320 KB WGP LDS)
    __shared__ float As[M_BLK * LDS_STRIDE];

    const int tid  = threadIdx.x;
    const int mblk = blockIdx.x;   // which 64-token tile
    const int nblk = blockIdx.y;   // which 256-column tile

    // ---- cooperative gather of 64 token rows of A into LDS via async DMA ----
    // GLOBAL_LOAD_ASYNC_TO_LDS_B128: per-lane 16 B copy from global memory to LDS, ASYNCcnt-
    // tracked, no VGPR round-trip (CDNA5 ISA).
    {
        const int row  = tid >> 2;          // 0..63
        const int part = tid & 3;           // 4 threads per row, 64 floats each
        const int tok  = x[mblk * M_BLK + row];
        unsigned long long ga =
            (unsigned long long)(A + (long)tok * RANK + part * 64);
        // Low 32 bits of a generic shared-aperture address == LDS byte offset.
        unsigned lo =
            (unsigned)(unsigned long long)(As + row * LDS_STRIDE + part * 64);
#pragma unroll
        for (int i = 0; i < 16; ++i) {
            asm volatile("global_load_async_to_lds_b128 %0, %1, off"
                         :: "v"(lo), "v"(ga) : "memory");
            lo += 16;
            ga += 16;
        }
        asm volatile("s_wait_asynccnt 0x0" ::: "memory");
    }
    __syncthreads();

    // ---- wave tiling: 8 waves as 2 (M) x 4 (N); each wave: M=32, N=64 ----
    const int lane   = tid & 31;
    const int wave   = tid >> 5;
    const int wave_m = wave >> 2;   // 0..1 -> 32 tokens
    const int wave_n = wave & 3;    // 0..3 -> 64 columns
    const int half   = lane >> 4;   // K-half select per WMMA f32 A/B layout
    const int l16    = lane & 15;   // M (for A) / N (for B) index within tile

    v8f acc[2][4] = {};

    // Per-lane column base for B fragments (lane l16 owns column n_base + l16)
    const float* Bp  = B + (long)(nblk * N_BLK + wave_n * 64 + l16);
    const float* As0 = As + (wave_m * 32 +  0 + l16) * LDS_STRIDE;
    const float* As1 = As + (wave_m * 32 + 16 + l16) * LDS_STRIDE;

#pragma unroll 2
    for (int k = 0; k < RANK; k += 4) {
        const int kk = k + 2 * half;           // lanes 0-15: K=k,k+1; 16-31: K=k+2,k+3
        // A fragments (ds_load_b64 pairs, bank-conflict free via stride padding)
        v2f a0 = *(const v2f*)(As0 + kk);
        v2f a1 = *(const v2f*)(As1 + kk);
#pragma unroll
        for (int j = 0; j < 4; ++j) {
            v2f b;
            b.x = Bp[(long)(kk)     * EMB_DIM + j * 16];
            b.y = Bp[(long)(kk + 1) * EMB_DIM + j * 16];
            acc[0][j] = __builtin_amdgcn_wmma_f32_16x16x4_f32(
                false, a0, false, b, (short)0, acc[0][j], false, false);
            acc[1][j] = __builtin_amdgcn_wmma_f32_16x16x4_f32(
                false, a1, false, b, (short)0, acc[1][j], false, false);
        }
    }

    // ---- store: C/D layout: VGPR r -> M = r (lanes 0-15) / r+8 (lanes 16-31) ----
    const int col = nblk * N_BLK + wave_n * 64 + l16;
#pragma unroll
    for (int ms = 0; ms < 2; ++ms) {
#pragma unroll
        for (int j = 0; j < 4; ++j) {
#pragma unroll
            for (int r = 0; r < 8; ++r) {
                const int tokrow = mblk * M_BLK + wave_m * 32 + ms * 16 + r + 8 * half;
                // NT store: 64 MB output stream must not evict B/A from L2
                __builtin_nontemporal_store(
                    acc[ms][j][r], out + (long)tokrow * EMB_DIM + col + j * 16);
            }
        }
    }
}

extern "C" void kernel_launch(void* const* d_in, const int* in_sizes, int n_in,
                              void* d_out, int out_size, void* d_ws, size_t ws_size,
                              hipStream_t stream) {
    const int*   x   = (const int*)d_in[0];    // [8, 2048] token ids
    const float* A   = (const float*)d_in[1];  // [50257, 256]
    const float* Bw  = (const float*)d_in[2];  // [256, 1024]
    float*       out = (float*)d_out;          // [8, 2048, 1024]

    dim3 grid(NUM_TOKENS / M_BLK, EMB_DIM / N_BLK);  // (256, 4)
    factored_embed_wmma<<<grid, dim3(256), 0, stream>>>(x, A, Bw, out);
}